// PBRModel_79834852098660
// MI455X (gfx1250) — compile-verified
//
#include <hip/hip_runtime.h>
#include <hip/hip_bf16.h>

// PBR image-based-lighting shade pass for MI455X (gfx1250).
// Bandwidth-bound: ~369 MB streamed per launch vs 23.3 TB/s HBM (~16 us floor).
// Strategy: NT cache hints on the one-touch per-pixel streams so the two 18 MB
// cubemaps + FG LUT stay resident in the 192 MB L2 for the random bilinear
// gathers; global_prefetch_b8 to run ahead of the gather latency; wave32
// one-pixel-per-lane mapping. No matmul structure -> WMMA intentionally unused.

typedef __attribute__((ext_vector_type(4))) float f32x4;
typedef __attribute__((ext_vector_type(2))) float f32x2;

#define PBR_EPS 1e-12f

struct Vec3 { float x, y, z; };

__device__ __forceinline__ int iclampi(int v, int lo, int hi) {
  return v < lo ? lo : (v > hi ? hi : v);
}

__device__ __forceinline__ Vec3 load_vec3_nt(const float* __restrict__ base, int i) {
  const float* p = base + (size_t)i * 3;
  Vec3 v;
  v.x = __builtin_nontemporal_load(p + 0);
  v.y = __builtin_nontemporal_load(p + 1);
  v.z = __builtin_nontemporal_load(p + 2);
  return v;
}

__device__ __forceinline__ void store_vec3_nt(float* __restrict__ base, int i,
                                              float a, float b, float c) {
  float* p = base + (size_t)i * 3;
  __builtin_nontemporal_store(a, p + 0);
  __builtin_nontemporal_store(b, p + 1);
  __builtin_nontemporal_store(c, p + 2);
}

__device__ __forceinline__ Vec3 normalize3(Vec3 v) {
  float n = sqrtf(v.x * v.x + v.y * v.y + v.z * v.z);
  float inv = 1.0f / fmaxf(n, PBR_EPS);
  Vec3 r; r.x = v.x * inv; r.y = v.y * inv; r.z = v.z * inv;
  return r;
}

// Bilinear cubemap sample, per-face edge-clamped. cube: [6,512,512,3] f32.
// Gathers use default (RT) temporal policy -> serviced from L2 (18 MB << 192 MB L2).
__device__ __forceinline__ void sample_cube512(const float* __restrict__ cube,
                                               float vx, float vy, float vz,
                                               float out[3]) {
  const int R = 512;
  float ax = fabsf(vx), ay = fabsf(vy), az = fabsf(vz);
  bool is_x = (ax >= ay) && (ax >= az);
  bool is_y = (!is_x) && (ay >= az);
  float ma = fmaxf(is_x ? ax : (is_y ? ay : az), PBR_EPS);
  float sc = is_x ? (vx >= 0.0f ? -vz : vz)
                  : (is_y ? vx : (vz >= 0.0f ? vx : -vx));
  float tc = is_x ? -vy
                  : (is_y ? (vy >= 0.0f ? vz : -vz) : -vy);
  int face = is_x ? (vx >= 0.0f ? 0 : 1)
                  : (is_y ? (vy >= 0.0f ? 2 : 3) : (vz >= 0.0f ? 4 : 5));
  float u = (sc / ma + 1.0f) * 0.5f;
  float w = (tc / ma + 1.0f) * 0.5f;
  float xf = u * (float)R - 0.5f;
  float yf = w * (float)R - 0.5f;
  float x0 = floorf(xf), y0 = floorf(yf);
  float fx = xf - x0, fy = yf - y0;
  int x0i = iclampi((int)x0,     0, R - 1);
  int x1i = iclampi((int)x0 + 1, 0, R - 1);
  int y0i = iclampi((int)y0,     0, R - 1);
  int y1i = iclampi((int)y0 + 1, 0, R - 1);
  int rowbase = face * R;
  const float* __restrict__ p00 = cube + (size_t)((rowbase + y0i) * R + x0i) * 3;
  const float* __restrict__ p01 = cube + (size_t)((rowbase + y0i) * R + x1i) * 3;
  const float* __restrict__ p10 = cube + (size_t)((rowbase + y1i) * R + x0i) * 3;
  const float* __restrict__ p11 = cube + (size_t)((rowbase + y1i) * R + x1i) * 3;
  float omfx = 1.0f - fx, omfy = 1.0f - fy;
#pragma unroll
  for (int c = 0; c < 3; ++c) {
    float top = p00[c] * omfx + p01[c] * fx;
    float bot = p10[c] * omfx + p11[c] * fx;
    out[c] = top * omfy + bot * fy;
  }
}

// Bilinear 2D sample, clamp boundary. tex: [256,256,2] f32 (0.5 MB, L2-resident).
__device__ __forceinline__ void sample_fg256(const float* __restrict__ tex,
                                             float u, float v, float out[2]) {
  const int W = 256, H = 256;
  float xf = u * (float)W - 0.5f;
  float yf = v * (float)H - 0.5f;
  float x0 = floorf(xf), y0 = floorf(yf);
  float fx = xf - x0, fy = yf - y0;
  int x0i = iclampi((int)x0,     0, W - 1);
  int x1i = iclampi((int)x0 + 1, 0, W - 1);
  int y0i = iclampi((int)y0,     0, H - 1);
  int y1i = iclampi((int)y0 + 1, 0, H - 1);
  const f32x2* __restrict__ t = (const f32x2*)tex;
  f32x2 c00 = t[y0i * W + x0i];
  f32x2 c01 = t[y0i * W + x1i];
  f32x2 c10 = t[y1i * W + x0i];
  f32x2 c11 = t[y1i * W + x1i];
  f32x2 top = c00 * (1.0f - fx) + c01 * fx;
  f32x2 bot = c10 * (1.0f - fx) + c11 * fx;
  f32x2 r   = top * (1.0f - fy) + bot * fy;
  out[0] = r.x;
  out[1] = r.y;
}

__global__ void __launch_bounds__(256)
pbr_shade_kernel(const float* __restrict__ view_position,
                 const float* __restrict__ world_position,
                 const float* __restrict__ world_normal,
                 const float* __restrict__ map_Kd,
                 const float* __restrict__ map_Ks,
                 const float* __restrict__ light_diffuse,
                 const float* __restrict__ light_specular,
                 const float* __restrict__ fg_lut,
                 float* __restrict__ out_diffuse,
                 float* __restrict__ out_specular,
                 int n) {
  int i = blockIdx.x * blockDim.x + threadIdx.x;
  if (i >= n) return;

  // Run-ahead prefetch of the streaming inputs (~4K pixels ahead) to cover
  // HBM latency while the gather-heavy math executes. Speculative, harmless
  // near the tail. Lowers to global_prefetch_b8 on gfx1250.
  const int PF = 4096;
  if (i + PF < n) {
    __builtin_prefetch(view_position  + (size_t)(i + PF) * 3, 0, 0);
    __builtin_prefetch(world_position + (size_t)(i + PF) * 3, 0, 0);
    __builtin_prefetch(world_normal   + (size_t)(i + PF) * 3, 0, 0);
    __builtin_prefetch(map_Kd         + (size_t)(i + PF) * 4, 0, 0);
    __builtin_prefetch(map_Ks         + (size_t)(i + PF) * 3, 0, 0);
  }

  // One-touch streams: non-temporal loads so they bypass/evict-first in L2,
  // keeping the cubemaps + LUT resident for the random gathers.
  Vec3 vp = load_vec3_nt(view_position, i);
  Vec3 wp = load_vec3_nt(world_position, i);
  Vec3 wn = load_vec3_nt(world_normal, i);
  f32x4 kd = __builtin_nontemporal_load((const f32x4*)map_Kd + i);
  Vec3 ks = load_vec3_nt(map_Ks, i);

  Vec3 nrm = normalize3(wn);
  Vec3 d;
  d.x = vp.x - wp.x; d.y = vp.y - wp.y; d.z = vp.z - wp.z;
  Vec3 wo = normalize3(d);

  float dwn = wo.x * nrm.x + wo.y * nrm.y + wo.z * nrm.z;
  Vec3 refl;
  refl.x = 2.0f * dwn * nrm.x - wo.x;
  refl.y = 2.0f * dwn * nrm.y - wo.y;
  refl.z = 2.0f * dwn * nrm.z - wo.z;
  Vec3 wi = normalize3(refl);

  float cos_wo_nrm = fminf(fmaxf(dwn, 0.0f), 1.0f);

  float albedo[3] = {kd.x, kd.y, kd.z};
  float rough = ks.y;   // map_Ks[...,1]
  float metal = ks.z;   // map_Ks[...,2]
  // map_specular is the constant 0.5 -> (1 - map_specular) == 0.5.

  float dl[3], sl[3], fg[2];
  sample_cube512(light_diffuse,  nrm.x, nrm.y, nrm.z, dl);
  sample_cube512(light_specular, wi.x,  wi.y,  wi.z,  sl);
  sample_fg256(fg_lut, cos_wo_nrm, rough, fg);

  float one_m_metal = 1.0f - metal;
  float od[3], os[3];
#pragma unroll
  for (int c = 0; c < 3; ++c) {
    float coef_d = albedo[c] * one_m_metal;
    float coef_s = (0.04f * one_m_metal + albedo[c] * metal) * 0.5f;
    od[c] = coef_d * dl[c];
    os[c] = (coef_s * fg[0] + fg[1]) * sl[c];
  }

  // Write-once outputs: non-temporal stores (streaming, no read-back).
  store_vec3_nt(out_diffuse,  i, od[0], od[1], od[2]);
  store_vec3_nt(out_specular, i, os[0], os[1], os[2]);
}

extern "C" void kernel_launch(void* const* d_in, const int* in_sizes, int n_in,
                              void* d_out, int out_size, void* d_ws, size_t ws_size,
                              hipStream_t stream) {
  const float* view_position  = (const float*)d_in[0];
  const float* world_position = (const float*)d_in[1];
  const float* world_normal   = (const float*)d_in[2];
  const float* map_Kd         = (const float*)d_in[3];
  const float* map_Ks         = (const float*)d_in[4];
  const float* light_diffuse  = (const float*)d_in[5];
  const float* light_specular = (const float*)d_in[6];
  const float* fg_lut         = (const float*)d_in[7];

  int n = in_sizes[0] / 3;  // number of pixels (B*H*W)

  float* out_diffuse  = (float*)d_out;
  float* out_specular = out_diffuse + (size_t)n * 3;

  const int block = 256;  // 8 wave32 waves per block
  const int grid = (n + block - 1) / block;
  pbr_shade_kernel<<<grid, block, 0, stream>>>(
      view_position, world_position, world_normal, map_Kd, map_Ks,
      light_diffuse, light_specular, fg_lut, out_diffuse, out_specular, n);
}